// SAGEReranker_48885317763303
// MI455X (gfx1250) — compile-verified
//
#include <hip/hip_runtime.h>
#include <math.h>

typedef __attribute__((ext_vector_type(2))) float v2f;
typedef __attribute__((ext_vector_type(8))) float v8f;

#define DIM 128
#define ROWT 5                  // row tiles (of 16 nodes) per wave -> 80 nodes/block
#define NODES_PER_BLOCK (ROWT * 16)

// ---------------------------------------------------------------------------
// Degree: one f32 atomic per edge
// ---------------------------------------------------------------------------
__global__ void sage_deg_kernel(const int* __restrict__ dst, float* __restrict__ deg, int E) {
    int e = blockIdx.x * blockDim.x + threadIdx.x;
    if (e < E) atomicAdd(&deg[dst[e]], 1.0f);
}

// ---------------------------------------------------------------------------
// Scatter-add: one wave per edge, lane q moves 4 contiguous floats.
// x/h and agg are L2-resident (51 MB each vs 192 MB L2), so this runs at L2
// atomic bandwidth; 128 consecutive f32 atomics per edge coalesce well.
// ---------------------------------------------------------------------------
__global__ void __launch_bounds__(256)
sage_scatter_kernel(const float* __restrict__ h,
                    const int* __restrict__ src,
                    const int* __restrict__ dst,
                    float* __restrict__ agg, int E) {
    int t = blockIdx.x * blockDim.x + threadIdx.x;
    int e = t >> 5;
    int q = (t & 31) * 4;
    if (e < E) {
        int s = src[e];
        int d = dst[e];
        const float4 v = *(const float4*)(h + (size_t)s * DIM + q);
        float* ap = agg + (size_t)d * DIM + q;
        atomicAdd(ap + 0, v.x);
        atomicAdd(ap + 1, v.y);
        atomicAdd(ap + 2, v.z);
        atomicAdd(ap + 3, v.w);
    }
}

// ---------------------------------------------------------------------------
// Mean-normalize: agg[n,:] *= 1/max(deg[n],1), in place.
// One wave per node, float4 per lane: fully coalesced, ~100 MB of mostly
// L2-resident traffic. Hoisting this out of the GEMM removes the per-
// fragment v_pk_mul dependency and frees 10 VGPRs in the WMMA loop.
// ---------------------------------------------------------------------------
__global__ void __launch_bounds__(256)
sage_norm_kernel(float* __restrict__ agg, const float* __restrict__ deg, int N) {
    int t = blockIdx.x * blockDim.x + threadIdx.x;
    int n = t >> 5;
    int q = (t & 31) * 4;
    if (n < N) {
        const float invd = 1.0f / fmaxf(deg[n], 1.0f);
        float4* p = (float4*)(agg + (size_t)n * DIM + q);
        float4 v = *p;
        v.x *= invd; v.y *= invd; v.z *= invd; v.w *= invd;
        *p = v;
    }
}

// ---------------------------------------------------------------------------
// WMMA fp32 16x16x4 helper.
//   A 16x4  : lane row = lane&15, VGPR0 = A[row, k + 2*(lane>>4)], VGPR1 = +1
//   B 4x16  : lane col = lane&15, VGPR0 = B[k + 2*(lane>>4), col], VGPR1 = +1
//   C 16x16 : vgpr r -> (M = r + 8*(lane>>4), N = lane&15)
// ---------------------------------------------------------------------------
__device__ __forceinline__ v8f wmma_f32_k4(v2f a, v2f b, v8f c) {
    return __builtin_amdgcn_wmma_f32_16x16x4_f32(
        false, a, false, b, (short)0, c, false, false);
}

// ---------------------------------------------------------------------------
// SAGE layer GEMM core: for ROWT row tiles x 2 col tiles,
//   acc[rt][ct] += aggn @ Wl^T + X @ Wr^T  over K = DIM
// (aggn is already mean-normalized). B fragments loaded once per K-step and
// reused over ROWT row tiles (20 WMMAs / 14 loads per step); row-tile stride
// (8192 B) and unrolled K offsets fold into load immediates; loads feed the
// WMMAs directly (no VALU dep).
// ---------------------------------------------------------------------------
__device__ __forceinline__ void sage_gemm_core(
        const float* __restrict__ aggn, const float* __restrict__ x,
        const float* __restrict__ Wl, const float* __restrict__ Wr,
        int nodeBase, int colBase, int lane,
        v8f acc0[ROWT], v8f acc1[ROWT]) {
    const int row = lane & 15;
    const int khalf = (lane >> 4) * 2;       // 0 or 2
    const int col0 = colBase + (lane & 15);
    const int col1 = col0 + 16;

    // per-lane bases; rt adds rt*16*DIM floats (immediate 8192 B steps)
    const float* Ap = aggn + (size_t)(nodeBase + row) * DIM + khalf;
    const float* Xp = x    + (size_t)(nodeBase + row) * DIM + khalf;
    const float* Bl0 = Wl + (size_t)col0 * DIM + khalf;   // B[k,col] = Wl[col,k]
    const float* Bl1 = Wl + (size_t)col1 * DIM + khalf;
    const float* Br0 = Wr + (size_t)col0 * DIM + khalf;
    const float* Br1 = Wr + (size_t)col1 * DIM + khalf;

#pragma unroll 2
    for (int k = 0; k < DIM; k += 4) {
        v2f b0 = *(const v2f*)(Bl0 + k);
        v2f b1 = *(const v2f*)(Bl1 + k);
        v2f c0 = *(const v2f*)(Br0 + k);
        v2f c1 = *(const v2f*)(Br1 + k);
        v2f a[ROWT], ax[ROWT];
#pragma unroll
        for (int rt = 0; rt < ROWT; ++rt) {
            a[rt]  = *(const v2f*)(Ap + k + rt * (16 * DIM));
            ax[rt] = *(const v2f*)(Xp + k + rt * (16 * DIM));
        }
#pragma unroll
        for (int rt = 0; rt < ROWT; ++rt) {
            acc0[rt] = wmma_f32_k4(a[rt],  b0, acc0[rt]);
            acc1[rt] = wmma_f32_k4(a[rt],  b1, acc1[rt]);
            acc0[rt] = wmma_f32_k4(ax[rt], c0, acc0[rt]);
            acc1[rt] = wmma_f32_k4(ax[rt], c1, acc1[rt]);
        }
    }
}

// ---------------------------------------------------------------------------
// Layer-1: h1 = relu( aggn @ Wl^T + bl + x @ Wr^T )
// Block = 128 threads = 4 waves; block owns 80 nodes x 128 cols.
// ---------------------------------------------------------------------------
__global__ void __launch_bounds__(128)
__attribute__((amdgpu_waves_per_eu(1)))
sage_gemm1_kernel(const float* __restrict__ aggn, const float* __restrict__ x,
                  const float* __restrict__ Wl, const float* __restrict__ bl,
                  const float* __restrict__ Wr, float* __restrict__ out) {
    const int lane = threadIdx.x & 31;
    const int wave = threadIdx.x >> 5;           // 0..3
    const int nodeBase = blockIdx.x * NODES_PER_BLOCK;
    const int colBase = wave * 32;

    v8f acc0[ROWT] = {};
    v8f acc1[ROWT] = {};
    sage_gemm_core(aggn, x, Wl, Wr, nodeBase, colBase, lane, acc0, acc1);

    const int col0 = colBase + (lane & 15);
    const int col1 = col0 + 16;
    const float bias0 = bl[col0];
    const float bias1 = bl[col1];

#pragma unroll
    for (int rt = 0; rt < ROWT; ++rt) {
#pragma unroll
        for (int vi = 0; vi < 8; ++vi) {
            const int orow = nodeBase + rt * 16 + vi + (lane >> 4) * 8;
            out[(size_t)orow * DIM + col0] = fmaxf(acc0[rt][vi] + bias0, 0.0f);
            out[(size_t)orow * DIM + col1] = fmaxf(acc1[rt][vi] + bias1, 0.0f);
        }
    }
}

// ---------------------------------------------------------------------------
// Layer-2 fused with score head:
//   h2 = relu(aggn @ Wl2^T + bl2 + h1 @ Wr2^T)
//   gnn = h2 @ score_w^T + score_b
//   out = sig(alpha)*reranker + (1-sig(alpha))*gnn
// ---------------------------------------------------------------------------
__global__ void __launch_bounds__(128)
__attribute__((amdgpu_waves_per_eu(1)))
sage_gemm2_score_kernel(const float* __restrict__ aggn, const float* __restrict__ h1,
                        const float* __restrict__ Wl, const float* __restrict__ bl,
                        const float* __restrict__ Wr,
                        const float* __restrict__ score_w,
                        const float* __restrict__ score_b,
                        const float* __restrict__ rer,
                        const float* __restrict__ alpha_p,
                        float* __restrict__ out) {
    const int lane = threadIdx.x & 31;
    const int wave = threadIdx.x >> 5;
    const int nodeBase = blockIdx.x * NODES_PER_BLOCK;
    const int colBase = wave * 32;

    v8f acc0[ROWT] = {};
    v8f acc1[ROWT] = {};
    sage_gemm_core(aggn, h1, Wl, Wr, nodeBase, colBase, lane, acc0, acc1);

    const int col0 = colBase + (lane & 15);
    const int col1 = col0 + 16;
    const float bias0 = bl[col0];
    const float bias1 = bl[col1];
    const float sw0 = score_w[col0];
    const float sw1 = score_w[col1];

    __shared__ float partial[NODES_PER_BLOCK][4];

#pragma unroll
    for (int rt = 0; rt < ROWT; ++rt) {
#pragma unroll
        for (int vi = 0; vi < 8; ++vi) {
            float v0 = fmaxf(acc0[rt][vi] + bias0, 0.0f);
            float v1 = fmaxf(acc1[rt][vi] + bias1, 0.0f);
            float s = v0 * sw0 + v1 * sw1;       // dot over this wave's 32 cols
#pragma unroll
            for (int off = 1; off < 16; off <<= 1)
                s += __shfl_xor(s, off, 32);
            if ((lane & 15) == 0)
                partial[rt * 16 + vi + (lane >> 4) * 8][wave] = s;
        }
    }
    __syncthreads();

    if (threadIdx.x < NODES_PER_BLOCK) {
        const int r = threadIdx.x;
        float gnn = partial[r][0] + partial[r][1] + partial[r][2] + partial[r][3]
                    + score_b[0];
        const float al = alpha_p[0];
        const float a = 1.0f / (1.0f + expf(-al));
        const int n = nodeBase + r;
        out[n] = a * rer[n] + (1.0f - a) * gnn;
    }
}

// ---------------------------------------------------------------------------
// Launch
// ---------------------------------------------------------------------------
extern "C" void kernel_launch(void* const* d_in, const int* in_sizes, int n_in,
                              void* d_out, int out_size, void* d_ws, size_t ws_size,
                              hipStream_t stream) {
    const float* x       = (const float*)d_in[0];
    const int*   edge    = (const int*)d_in[1];
    const float* rer     = (const float*)d_in[2];
    const float* Wl1     = (const float*)d_in[3];
    const float* bl1     = (const float*)d_in[4];
    const float* Wr1     = (const float*)d_in[5];
    const float* Wl2     = (const float*)d_in[6];
    const float* bl2     = (const float*)d_in[7];
    const float* Wr2     = (const float*)d_in[8];
    const float* score_w = (const float*)d_in[9];
    const float* score_b = (const float*)d_in[10];
    const float* alpha   = (const float*)d_in[11];

    const int N = in_sizes[0] / DIM;
    const int E = in_sizes[1] / 2;
    const int* src = edge;
    const int* dst = edge + E;

    // workspace layout: deg [N] | agg [N*128] | h1 [N*128]
    float* deg = (float*)d_ws;
    float* agg = deg + N;
    float* h1  = agg + (size_t)N * DIM;

    float* out = (float*)d_out;

    const int scatterBlocks = (int)(((long long)E * 32 + 255) / 256);
    const int normBlocks    = (int)(((long long)N * 32 + 255) / 256);

    // zero deg + agg (contiguous)
    hipMemsetAsync(d_ws, 0, ((size_t)N + (size_t)N * DIM) * sizeof(float), stream);

    // degrees
    sage_deg_kernel<<<(E + 255) / 256, 256, 0, stream>>>(dst, deg, E);

    // layer 1: scatter x, normalize, WMMA gemm + relu
    sage_scatter_kernel<<<scatterBlocks, 256, 0, stream>>>(x, src, dst, agg, E);
    sage_norm_kernel<<<normBlocks, 256, 0, stream>>>(agg, deg, N);
    sage_gemm1_kernel<<<N / NODES_PER_BLOCK, 128, 0, stream>>>(agg, x, Wl1, bl1, Wr1, h1);

    // layer 2: re-zero agg, scatter h1, normalize, fused gemm + relu + score
    hipMemsetAsync(agg, 0, (size_t)N * DIM * sizeof(float), stream);
    sage_scatter_kernel<<<scatterBlocks, 256, 0, stream>>>(h1, src, dst, agg, E);
    sage_norm_kernel<<<normBlocks, 256, 0, stream>>>(agg, deg, N);
    sage_gemm2_score_kernel<<<N / NODES_PER_BLOCK, 128, 0, stream>>>(agg, h1, Wl2, bl2, Wr2,
                                                                     score_w, score_b, rer, alpha, out);
}